// SimpleGCNLayer_1769526526583
// MI455X (gfx1250) — compile-verified
//
#include <hip/hip_runtime.h>

typedef __attribute__((ext_vector_type(2))) float v2f;
typedef __attribute__((ext_vector_type(8))) float v8f;

// ---------------------------------------------------------------------------
// Phase 1: degree = 1 (self loop) + segment_sum(edge_weight over rows)
// ---------------------------------------------------------------------------
__global__ void gcn_deg_init(float* __restrict__ deg, int N) {
    int i = blockIdx.x * blockDim.x + threadIdx.x;
    if (i < N) deg[i] = 1.0f;  // self-loop weight
}

__global__ void gcn_deg_acc(const int* __restrict__ rowIdx,
                            const float* __restrict__ w,
                            float* __restrict__ deg, int E) {
    int e = blockIdx.x * blockDim.x + threadIdx.x;
    if (e < E) atomicAdd(&deg[rowIdx[e]], w[e]);
}

__global__ void gcn_dinv(const float* __restrict__ deg,
                         float* __restrict__ dinv, int N) {
    int i = blockIdx.x * blockDim.x + threadIdx.x;
    if (i < N) {
        float d = deg[i];
        dinv[i] = (d > 0.0f) ? rsqrtf(d) : 0.0f;
    }
}

// ---------------------------------------------------------------------------
// Phase 2: y = x @ W^T  via V_WMMA_F32_16X16X4_F32 (wave32, one 16x16 tile
// per wave, 4 waves per block cover N-dim 0..63, K=64 -> 16 chained WMMAs)
// ---------------------------------------------------------------------------
__global__ void __launch_bounds__(128)
gcn_gemm_xw(const float* __restrict__ x, const float* __restrict__ W,
            float* __restrict__ y, int N) {
    const int lane = threadIdx.x & 31;
    const int wave = threadIdx.x >> 5;   // 0..3 -> output-column tile
    const int m0   = blockIdx.x * 16;
    const int n0   = wave * 16;

    const int half = lane >> 4;          // 0: K pair {0,1}, 1: K pair {2,3}
    const int l15  = lane & 15;

    // A fragment: row m0+l15, K = k0 + 2*half + {0,1}   (clamped, EXEC all-1)
    int arow = m0 + l15;
    if (arow >= N) arow = N - 1;
    const float* __restrict__ arow_p = x + (size_t)arow * 64;
    // B fragment: B[k][n] = W[n][k]; n = n0+l15, same K pairing
    const float* __restrict__ brow_p = W + (size_t)(n0 + l15) * 64;

    v8f c = {};
#pragma unroll
    for (int k0 = 0; k0 < 64; k0 += 4) {
        const int kb = k0 + 2 * half;
        v2f a = *(const v2f*)(arow_p + kb);
        v2f b = *(const v2f*)(brow_p + kb);
        // (neg_a, A, neg_b, B, c_mod, C, reuse_a, reuse_b)
        c = __builtin_amdgcn_wmma_f32_16x16x4_f32(false, a, false, b,
                                                  (short)0, c, false, false);
    }

    // D layout: VGPR r -> row m0 + r + 8*half, col n0 + l15
#pragma unroll
    for (int r = 0; r < 8; ++r) {
        int m = m0 + r + 8 * half;
        if (m < N) y[(size_t)m * 64 + n0 + l15] = c[r];
    }
}

// ---------------------------------------------------------------------------
// Phase 3: out = dinv^2 * y + b  (self-loop term + bias, initializes d_out)
// ---------------------------------------------------------------------------
__global__ void gcn_out_init(const float* __restrict__ y,
                             const float* __restrict__ dinv,
                             const float* __restrict__ b,
                             float* __restrict__ out, int total) {
    int gid = blockIdx.x * blockDim.x + threadIdx.x;
    if (gid < total) {
        int i = gid >> 6, d = gid & 63;
        float di = dinv[i];
        out[gid] = di * di * y[gid] + b[d];
    }
}

// ---------------------------------------------------------------------------
// Phase 4: edge scatter. One wave per edge; each lane carries a float2 chunk
// of the 64-wide feature row. y gathers hit the 192MB L2 (y = 25.6 MB);
// accumulation via global_atomic_add_f32.
// ---------------------------------------------------------------------------
__global__ void __launch_bounds__(256)
gcn_scatter(const int* __restrict__ rowIdx, const int* __restrict__ colIdx,
            const float* __restrict__ w, const float* __restrict__ dinv,
            const float* __restrict__ y, float* __restrict__ out, int E) {
    const int lane = threadIdx.x & 31;
    const int e = blockIdx.x * 8 + (threadIdx.x >> 5);
    if (e >= E) return;
    const int r = rowIdx[e];
    const int c = colIdx[e];
    const float coeff = dinv[r] * w[e] * dinv[c];
    v2f xv = *(const v2f*)(y + (size_t)c * 64 + 2 * lane);
    float* dst = out + (size_t)r * 64 + 2 * lane;
    atomicAdd(dst,     coeff * xv.x);
    atomicAdd(dst + 1, coeff * xv.y);
}

// ---------------------------------------------------------------------------
extern "C" void kernel_launch(void* const* d_in, const int* in_sizes, int n_in,
                              void* d_out, int out_size, void* d_ws, size_t ws_size,
                              hipStream_t stream) {
    const float* x  = (const float*)d_in[0];
    const int*   ei = (const int*)d_in[1];    // [2, E] row-major: rows then cols
    const float* ew = (const float*)d_in[2];
    const float* W  = (const float*)d_in[3];  // [D_OUT, D_IN] row-major
    const float* b  = (const float*)d_in[4];
    float* out = (float*)d_out;

    const int N = in_sizes[0] / 64;
    const int E = in_sizes[1] / 2;
    const int* rowIdx = ei;
    const int* colIdx = ei + E;

    // workspace: deg[N] | dinv[N] | y[N*64]
    char* ws = (char*)d_ws;
    float* deg  = (float*)ws;
    float* dinv = deg + N;
    size_t yoff = (((size_t)2 * N * sizeof(float)) + 255) & ~(size_t)255;
    float* y    = (float*)(ws + yoff);

    gcn_deg_init<<<(N + 255) / 256, 256, 0, stream>>>(deg, N);
    gcn_deg_acc <<<(E + 255) / 256, 256, 0, stream>>>(rowIdx, ew, deg, E);
    gcn_dinv    <<<(N + 255) / 256, 256, 0, stream>>>(deg, dinv, N);
    gcn_gemm_xw <<<(N + 15) / 16, 128, 0, stream>>>(x, W, y, N);
    gcn_out_init<<<(N * 64 + 255) / 256, 256, 0, stream>>>(y, dinv, b, out, N * 64);
    gcn_scatter <<<(E + 7) / 8, 256, 0, stream>>>(rowIdx, colIdx, ew, dinv, y, out, E);
}